// AtomAttentionEncoderPairformer_2370821947558
// MI455X (gfx1250) — compile-verified
//
#include <hip/hip_runtime.h>
#include <hip/hip_bf16.h>

#define LATOMS 2048
#define NTOK   512
#define CATOM  128
#define CPAIR  16
#define CTOK   384
#define F1D    388
#define NBLK   3
#define NHEAD  4
#define HDIM   32
#define FFDIM  512

typedef __attribute__((ext_vector_type(2))) float v2f;
typedef __attribute__((ext_vector_type(8))) float v8f;

static __device__ __forceinline__ v8f wmma4(v2f a, v2f b, v8f c) {
    // D = A(16x4, f32) * B(4x16, f32) + C(16x16, f32), exact f32 semantics
    return __builtin_amdgcn_wmma_f32_16x16x4_f32(false, a, false, b, (short)0, c, false, false);
}
static __device__ __forceinline__ v8f vzero8() {
    v8f r; 
#pragma unroll
    for (int v = 0; v < 8; ++v) r[v] = 0.f;
    return r;
}

// ---------------------------------------------------------------------------
// Transpose a D-layout 16x16 tile through LDS and fetch A-layout fragments.
// Block-uniform call sites only (all waves execute identical sequence).
// ---------------------------------------------------------------------------
static __device__ __forceinline__ void transpose_stage(float* T, const v8f& src, bool dorelu,
                                                       int lane, v2f afr[4]) {
    const int n = lane & 15, half = lane >> 4, kh = half * 2;
    __syncthreads();
#pragma unroll
    for (int v = 0; v < 8; ++v) {
        float val = src[v];
        if (dorelu) val = fmaxf(val, 0.f);
        T[(v + 8 * half) * 16 + n] = val;
    }
    __syncthreads();
#pragma unroll
    for (int kc = 0; kc < 4; ++kc) {
        int k0 = kc * 4 + kh;
        afr[kc].x = T[n * 16 + k0];
        afr[kc].y = T[n * 16 + k0 + 1];
    }
}

// ---------------------------------------------------------------------------
// feat = concat(ref_pos, ref_charge, ref_element, ref_atom_name_chars)
// ---------------------------------------------------------------------------
__global__ void feat_kernel(const float* __restrict__ pos, const float* __restrict__ chg,
                            const float* __restrict__ elem, const float* __restrict__ chars,
                            float* __restrict__ feat) {
    int idx = blockIdx.x * 256 + threadIdx.x;
    if (idx >= LATOMS * F1D) return;
    int l = idx / F1D, c = idx - l * F1D;
    float v;
    if (c < 3)        v = pos[l * 3 + c];
    else if (c == 3)  v = chg[l];
    else if (c < 132) v = elem[l * 128 + (c - 4)];
    else              v = chars[l * 256 + (c - 132)];
    feat[idx] = v;
}

__global__ void copy_kernel(const float* __restrict__ src, float* __restrict__ dst, int n) {
    int i = blockIdx.x * 256 + threadIdx.x;
    if (i < n) dst[i] = src[i];
}

// ---------------------------------------------------------------------------
// Generic f32 WMMA GEMM: Out = [relu?](A[M,K] (relu'd?) @ B[K,N] + Cres?)
// One wave per 16x16 tile. K multiple of 4, M/N multiples of 16.
// ---------------------------------------------------------------------------
template <bool RELUA, bool RELUO, bool RESID>
__global__ __launch_bounds__(128) void gemm_kernel(const float* __restrict__ A,
                                                   const float* __restrict__ B,
                                                   const float* Cres, float* Out,
                                                   int M, int N, int K) {
    const int wave = threadIdx.x >> 5, lane = threadIdx.x & 31;
    const int tile = blockIdx.x * 4 + wave;
    const int tilesN = N >> 4;
    const int ntiles = (M >> 4) * tilesN;
    if (tile >= ntiles) return;
    const int r0 = (tile / tilesN) << 4;
    const int c0 = (tile - (tile / tilesN) * tilesN) << 4;
    const int n = lane & 15, kh = (lane >> 4) * 2;

    v8f acc = vzero8();
    const float* Arow = A + (size_t)(r0 + n) * K;
    for (int k0 = kh; k0 < K; k0 += 4) {
        v2f a;
        if (RELUA) { a.x = fmaxf(Arow[k0], 0.f); a.y = fmaxf(Arow[k0 + 1], 0.f); }
        else       { a = *(const v2f*)(Arow + k0); }
        v2f b;
        b.x = B[(size_t)k0 * N + c0 + n];
        b.y = B[(size_t)(k0 + 1) * N + c0 + n];
        acc = wmma4(a, b, acc);
    }
#pragma unroll
    for (int v = 0; v < 8; ++v) {
        int m = v + ((lane >> 4) << 3);
        float val = acc[v];
        if (RESID) val += Cres[(size_t)(r0 + m) * N + c0 + n];
        if (RELUO) val = fmaxf(val, 0.f);
        Out[(size_t)(r0 + m) * N + c0 + n] = val;
    }
}

// ---------------------------------------------------------------------------
// LayerNorm over 128 channels, one wave per row.
// ---------------------------------------------------------------------------
__global__ __launch_bounds__(256) void ln_kernel(const float* __restrict__ X,
                                                 float* __restrict__ Y) {
    const int wave = threadIdx.x >> 5, lane = threadIdx.x & 31;
    const int r = blockIdx.x * 8 + wave;
    if (r >= LATOMS) return;
    float4 x4 = *(const float4*)(X + (size_t)r * CATOM + lane * 4);
    float s = x4.x + x4.y + x4.z + x4.w;
#pragma unroll
    for (int d = 1; d < 32; d <<= 1) s += __shfl_xor(s, d);
    float mu = s * (1.f / 128.f);
    float d0 = x4.x - mu, d1 = x4.y - mu, d2 = x4.z - mu, d3 = x4.w - mu;
    float q = d0 * d0 + d1 * d1 + d2 * d2 + d3 * d3;
#pragma unroll
    for (int d = 1; d < 32; d <<= 1) q += __shfl_xor(q, d);
    float rs = rsqrtf(q * (1.f / 128.f) + 1e-5f);
    float4 y4 = {d0 * rs, d1 * rs, d2 * rs, d3 * rs};
    *(float4*)(Y + (size_t)r * CATOM + lane * 4) = y4;
}

// ---------------------------------------------------------------------------
// Pair kernel: per (i, j-tile of 16) compute geometry terms, pair MLP
// (residual), write P tile (contiguous 1KB) and all 12 bias planes.
// bias layout: [NBLK*NHEAD][L][L]
// ---------------------------------------------------------------------------
__global__ __launch_bounds__(128) void pair_kernel(
    const float* __restrict__ pos, const int* __restrict__ uid,
    const float* __restrict__ SL, const float* __restrict__ SM,
    const float* __restrict__ Wd, const float* __restrict__ Winvd,
    const float* __restrict__ Wvm, const float* __restrict__ Wp1,
    const float* __restrict__ Wp2, const float* __restrict__ Wp3,
    const float* __restrict__ Wb, float* __restrict__ Pout,
    float* __restrict__ biasws) {
    __shared__ float lds[4][256];
    const int wave = threadIdx.x >> 5, lane = threadIdx.x & 31;
    const int tile = blockIdx.x * 4 + wave;                 // grid sized exactly
    const int i  = tile >> 7;
    const int j0 = (tile & 127) << 4;
    const int n = lane & 15, half = lane >> 4, kh = half * 2;
    float* T = lds[wave];

    // Preload B-layout fragments of the 16x16 weights (and padded 16x12 Wb)
    v2f wp1f[4], wp2f[4], wp3f[4], wbf[4];
#pragma unroll
    for (int kc = 0; kc < 4; ++kc) {
        int k0 = kc * 4 + kh;
        wp1f[kc].x = Wp1[k0 * 16 + n];  wp1f[kc].y = Wp1[(k0 + 1) * 16 + n];
        wp2f[kc].x = Wp2[k0 * 16 + n];  wp2f[kc].y = Wp2[(k0 + 1) * 16 + n];
        wp3f[kc].x = Wp3[k0 * 16 + n];  wp3f[kc].y = Wp3[(k0 + 1) * 16 + n];
        float b0 = 0.f, b1 = 0.f;
        if (n < 12) {  // column n = blk*4 + head ;  Wb[blk][k][head]
            int bb = n >> 2, hh = n & 3;
            b0 = Wb[((size_t)bb * 16 + k0) * 4 + hh];
            b1 = Wb[((size_t)bb * 16 + k0 + 1) * 4 + hh];
        }
        wbf[kc].x = b0; wbf[kc].y = b1;
    }

    // Elementwise init of P0 (D-layout: row m = j offset, col n = channel)
    const float px = pos[i * 3 + 0], py = pos[i * 3 + 1], pz = pos[i * 3 + 2];
    const int   ui = uid[i];
    const float wd0 = Wd[0 * 16 + n], wd1 = Wd[1 * 16 + n], wd2 = Wd[2 * 16 + n];
    const float wiv = Winvd[n], wvm = Wvm[n], sli = SL[i * 16 + n];
    v8f P0;
#pragma unroll
    for (int v = 0; v < 8; ++v) {
        int m = v + 8 * half, j = j0 + m;
        float dx = px - pos[j * 3 + 0], dy = py - pos[j * 3 + 1], dz = pz - pos[j * 3 + 2];
        float vf = (ui == uid[j]) ? 1.f : 0.f;
        float invd = 1.f / (1.f + sqrtf(dx * dx + dy * dy + dz * dz));
        P0[v] = (dx * wd0 + dy * wd1 + dz * wd2) * vf + invd * wiv * vf + wvm * vf
                + sli + SM[(size_t)j * 16 + n];
    }

    v2f afr[4];
    // u1 = relu(relu(P0) @ Wp1)
    transpose_stage(T, P0, true, lane, afr);
    v8f u1 = vzero8();
#pragma unroll
    for (int kc = 0; kc < 4; ++kc) u1 = wmma4(afr[kc], wp1f[kc], u1);
    // u2 = relu(u1 @ Wp2)
    transpose_stage(T, u1, true, lane, afr);
    v8f u2 = vzero8();
#pragma unroll
    for (int kc = 0; kc < 4; ++kc) u2 = wmma4(afr[kc], wp2f[kc], u2);
    // Pf = P0 + relu(u2) @ Wp3
    transpose_stage(T, u2, true, lane, afr);
    v8f Pf = P0;
#pragma unroll
    for (int kc = 0; kc < 4; ++kc) Pf = wmma4(afr[kc], wp3f[kc], Pf);
    // bias = Pf @ Wb_all  (also leaves Pf staged in LDS for coalesced store)
    transpose_stage(T, Pf, false, lane, afr);
    v8f bi = vzero8();
#pragma unroll
    for (int kc = 0; kc < 4; ++kc) bi = wmma4(afr[kc], wbf[kc], bi);

    // Coalesced P tile store: contiguous 256 floats at P[i][j0..j0+15][:]
    float4* dstP = (float4*)(Pout + ((size_t)i * LATOMS + j0) * CPAIR);
    const float4* srcT = (const float4*)T;
    dstP[lane * 2 + 0] = srcT[lane * 2 + 0];
    dstP[lane * 2 + 1] = srcT[lane * 2 + 1];

    // Bias planes: plane index == column n (= blk*4+head) for n < 12
    if (n < 12) {
        float* bp = biasws + (size_t)n * LATOMS * LATOMS + (size_t)i * LATOMS + j0 + 8 * half;
        float4 lo = {bi[0], bi[1], bi[2], bi[3]};
        float4 hi = {bi[4], bi[5], bi[6], bi[7]};
        ((float4*)bp)[0] = lo;
        ((float4*)bp)[1] = hi;
    }
}

// ---------------------------------------------------------------------------
// Flash attention: one wave per (query-tile, head). 16x16 S tiles via WMMA,
// online softmax (shfl row reductions within 16-lane halves), PV via WMMA.
// ---------------------------------------------------------------------------
__global__ __launch_bounds__(128) void attn_kernel(
    const float* __restrict__ qm, const float* __restrict__ km,
    const float* __restrict__ vm, const float* __restrict__ biasws,
    float* __restrict__ om, int bblk) {
    __shared__ float lds[4][256];
    const int wave = threadIdx.x >> 5, lane = threadIdx.x & 31;
    const int wid = blockIdx.x * 4 + wave;  // 512 waves exactly
    const int h = wid & 3;
    const int i0 = (wid >> 2) << 4;
    const int n = lane & 15, half = lane >> 4, kh = half * 2;
    float* T = lds[wave];
    const float scale = 0.17677669529663687f;  // 1/sqrt(32)

    // A-layout fragments of q tile, pre-scaled
    v2f aq[8];
#pragma unroll
    for (int kc = 0; kc < 8; ++kc) {
        int k0 = kc * 4 + kh;
        const float* qp = qm + (size_t)(i0 + n) * CATOM + h * HDIM + k0;
        aq[kc].x = qp[0] * scale;
        aq[kc].y = qp[1] * scale;
    }

    v8f o0 = vzero8(), o1 = vzero8();
    float rmax[8], rl[8];
#pragma unroll
    for (int v = 0; v < 8; ++v) { rmax[v] = -1e30f; rl[v] = 0.f; }

    const float* bplane = biasws + (size_t)(bblk * NHEAD + h) * LATOMS * LATOMS + (size_t)i0 * LATOMS;

    for (int j0 = 0; j0 < LATOMS; j0 += 16) {
        // S = q @ k^T (scaled) + bias
        v8f S = vzero8();
#pragma unroll
        for (int kc = 0; kc < 8; ++kc) {
            int k0 = kc * 4 + kh;
            const float* kp = km + (size_t)(j0 + n) * CATOM + h * HDIM + k0;
            v2f bk; bk.x = kp[0]; bk.y = kp[1];
            S = wmma4(aq[kc], bk, S);
        }
#pragma unroll
        for (int v = 0; v < 8; ++v) {
            int m = v + 8 * half;
            S[v] += bplane[(size_t)m * LATOMS + j0 + n];
        }
        // online softmax update
        float corr[8];
        v8f p;
#pragma unroll
        for (int v = 0; v < 8; ++v) {
            float t = S[v];
#pragma unroll
            for (int d = 1; d < 16; d <<= 1) t = fmaxf(t, __shfl_xor(t, d));
            float newm = fmaxf(rmax[v], t);
            corr[v] = __expf(rmax[v] - newm);
            rmax[v] = newm;
            p[v] = __expf(S[v] - newm);
            float s = p[v];
#pragma unroll
            for (int d = 1; d < 16; d <<= 1) s += __shfl_xor(s, d);
            rl[v] = rl[v] * corr[v] + s;
            o0[v] *= corr[v];
            o1[v] *= corr[v];
        }
        // transpose p and accumulate o += p @ v_tile (two 16-wide dim chunks)
        __syncthreads();
#pragma unroll
        for (int v = 0; v < 8; ++v) T[(v + 8 * half) * 16 + n] = p[v];
        __syncthreads();
#pragma unroll
        for (int kc = 0; kc < 4; ++kc) {
            int k0 = kc * 4 + kh;
            v2f ap; ap.x = T[n * 16 + k0]; ap.y = T[n * 16 + k0 + 1];
            const float* vr0 = vm + (size_t)(j0 + k0) * CATOM + h * HDIM;
            const float* vr1 = vm + (size_t)(j0 + k0 + 1) * CATOM + h * HDIM;
            v2f b0; b0.x = vr0[n];      b0.y = vr1[n];
            v2f b1; b1.x = vr0[16 + n]; b1.y = vr1[16 + n];
            o0 = wmma4(ap, b0, o0);
            o1 = wmma4(ap, b1, o1);
        }
    }
#pragma unroll
    for (int v = 0; v < 8; ++v) {
        int m = v + 8 * half;
        float inv = 1.f / rl[v];
        om[(size_t)(i0 + m) * CATOM + h * HDIM + n]      = o0[v] * inv;
        om[(size_t)(i0 + m) * CATOM + h * HDIM + 16 + n] = o1[v] * inv;
    }
}

// ---------------------------------------------------------------------------
// Deterministic segment mean (tokens sorted): precompute ranges, then sum.
// ---------------------------------------------------------------------------
__global__ void tokrange_kernel(const int* __restrict__ tok, int* __restrict__ ranges) {
    int t = blockIdx.x * 256 + threadIdx.x;
    if (t >= NTOK) return;
    int lo = 0, hi = LATOMS;
    while (lo < hi) { int mid = (lo + hi) >> 1; if (tok[mid] < t) lo = mid + 1; else hi = mid; }
    int s = lo;
    lo = s; hi = LATOMS;
    while (lo < hi) { int mid = (lo + hi) >> 1; if (tok[mid] < t + 1) lo = mid + 1; else hi = mid; }
    ranges[2 * t] = s;
    ranges[2 * t + 1] = lo;
}

__global__ void segmean_kernel(const float* __restrict__ pq, const int* __restrict__ ranges,
                               float* __restrict__ AI) {
    int idx = blockIdx.x * 256 + threadIdx.x;
    if (idx >= NTOK * CTOK) return;
    int t = idx / CTOK, c = idx - t * CTOK;
    int s = ranges[2 * t], e = ranges[2 * t + 1];
    float sum = 0.f;
    for (int l = s; l < e; ++l) sum += pq[(size_t)l * CTOK + c];
    int cnt = e - s;
    AI[idx] = sum / (float)(cnt > 0 ? cnt : 1);
}

// ---------------------------------------------------------------------------
static inline int gemm_blocks(int M, int N) { int t = (M >> 4) * (N >> 4); return (t + 3) / 4; }

extern "C" void kernel_launch(void* const* d_in, const int* in_sizes, int n_in,
                              void* d_out, int out_size, void* d_ws, size_t ws_size,
                              hipStream_t stream) {
    const float* ref_pos   = (const float*)d_in[0];
    const float* ref_chg   = (const float*)d_in[1];
    const float* ref_elem  = (const float*)d_in[2];
    const float* ref_chars = (const float*)d_in[3];
    const int*   uid       = (const int*)d_in[4];
    const int*   tok       = (const int*)d_in[5];
    const float* W_in  = (const float*)d_in[6];
    const float* W_d   = (const float*)d_in[7];
    const float* W_invd= (const float*)d_in[8];
    const float* W_vm  = (const float*)d_in[9];
    const float* W_sl  = (const float*)d_in[10];
    const float* W_sm  = (const float*)d_in[11];
    const float* Wp1   = (const float*)d_in[12];
    const float* Wp2   = (const float*)d_in[13];
    const float* Wp3   = (const float*)d_in[14];
    const float* Wq_tok= (const float*)d_in[15];
    const float* Wq    = (const float*)d_in[16];
    const float* Wk    = (const float*)d_in[17];
    const float* Wv    = (const float*)d_in[18];
    const float* Wb    = (const float*)d_in[19];
    const float* Wo    = (const float*)d_in[20];
    const float* Wt1   = (const float*)d_in[21];
    const float* Wt2   = (const float*)d_in[22];

    float* out  = (float*)d_out;
    float* outAI = out;                                    // [512,384]
    float* outQ  = outAI + (size_t)NTOK * CTOK;            // [2048,128]
    float* outCL = outQ + (size_t)LATOMS * CATOM;          // [2048,128]
    float* outP  = outCL + (size_t)LATOMS * CATOM;         // [2048,2048,16]

    float* ws = (float*)d_ws;
    size_t off = 0;
    float* biasws = ws + off; off += (size_t)NBLK * NHEAD * LATOMS * LATOMS; // 192 MB
    float* featws = ws + off; off += (size_t)LATOMS * F1D;
    float* SLws   = ws + off; off += (size_t)LATOMS * CPAIR;
    float* SMws   = ws + off; off += (size_t)LATOMS * CPAIR;
    float* Qws    = ws + off; off += (size_t)LATOMS * CATOM;
    float* Xws    = ws + off; off += (size_t)LATOMS * CATOM;
    float* qws    = ws + off; off += (size_t)LATOMS * CATOM;
    float* kws    = ws + off; off += (size_t)LATOMS * CATOM;
    float* vws    = ws + off; off += (size_t)LATOMS * CATOM;
    float* ows    = ws + off; off += (size_t)LATOMS * CATOM;
    float* h1ws   = ws + off; off += (size_t)LATOMS * FFDIM;
    float* pqws   = ws + off; off += (size_t)LATOMS * CTOK;
    int*   ranges = (int*)(ws + off); off += 2 * NTOK;

    // 1. feat concat + C_L = feat @ W_in  (written straight into d_out)
    feat_kernel<<<(LATOMS * F1D + 255) / 256, 256, 0, stream>>>(ref_pos, ref_chg, ref_elem,
                                                                ref_chars, featws);
    gemm_kernel<false, false, false><<<gemm_blocks(LATOMS, CATOM), 128, 0, stream>>>(
        featws, W_in, nullptr, outCL, LATOMS, CATOM, F1D);

    // 2. SL = relu(C_L) @ W_sl, SM = relu(C_L) @ W_sm
    gemm_kernel<true, false, false><<<gemm_blocks(LATOMS, CPAIR), 128, 0, stream>>>(
        outCL, W_sl, nullptr, SLws, LATOMS, CPAIR, CATOM);
    gemm_kernel<true, false, false><<<gemm_blocks(LATOMS, CPAIR), 128, 0, stream>>>(
        outCL, W_sm, nullptr, SMws, LATOMS, CPAIR, CATOM);

    // 3. Q = C_L
    copy_kernel<<<(LATOMS * CATOM + 255) / 256, 256, 0, stream>>>(outCL, Qws, LATOMS * CATOM);

    // 4. Pair tensor + fused bias projection for all 3 blocks
    pair_kernel<<<(LATOMS * (LATOMS / 16)) / 4, 128, 0, stream>>>(
        ref_pos, uid, SLws, SMws, W_d, W_invd, W_vm, Wp1, Wp2, Wp3, Wb, outP, biasws);

    // 5. Atom transformer blocks
    for (int b = 0; b < NBLK; ++b) {
        ln_kernel<<<LATOMS / 8, 256, 0, stream>>>(Qws, Xws);
        gemm_kernel<false, false, false><<<gemm_blocks(LATOMS, CATOM), 128, 0, stream>>>(
            Xws, Wq + (size_t)b * CATOM * CATOM, nullptr, qws, LATOMS, CATOM, CATOM);
        gemm_kernel<false, false, false><<<gemm_blocks(LATOMS, CATOM), 128, 0, stream>>>(
            Xws, Wk + (size_t)b * CATOM * CATOM, nullptr, kws, LATOMS, CATOM, CATOM);
        gemm_kernel<false, false, false><<<gemm_blocks(LATOMS, CATOM), 128, 0, stream>>>(
            Xws, Wv + (size_t)b * CATOM * CATOM, nullptr, vws, LATOMS, CATOM, CATOM);
        attn_kernel<<<(LATOMS / 16) * NHEAD / 4, 128, 0, stream>>>(qws, kws, vws, biasws, ows, b);
        gemm_kernel<false, false, true><<<gemm_blocks(LATOMS, CATOM), 128, 0, stream>>>(
            ows, Wo + (size_t)b * CATOM * CATOM, Qws, Qws, LATOMS, CATOM, CATOM);
        ln_kernel<<<LATOMS / 8, 256, 0, stream>>>(Qws, Xws);
        gemm_kernel<false, true, false><<<gemm_blocks(LATOMS, FFDIM), 128, 0, stream>>>(
            Xws, Wt1 + (size_t)b * CATOM * FFDIM, nullptr, h1ws, LATOMS, FFDIM, CATOM);
        gemm_kernel<false, false, true><<<gemm_blocks(LATOMS, CATOM), 128, 0, stream>>>(
            h1ws, Wt2 + (size_t)b * FFDIM * CATOM, Qws, Qws, LATOMS, CATOM, FFDIM);
    }

    // 6. Outputs: Q, pQ = relu(Q @ Wq_tok), segment mean -> A_I
    copy_kernel<<<(LATOMS * CATOM + 255) / 256, 256, 0, stream>>>(Qws, outQ, LATOMS * CATOM);
    gemm_kernel<false, true, false><<<gemm_blocks(LATOMS, CTOK), 128, 0, stream>>>(
        Qws, Wq_tok, nullptr, pqws, LATOMS, CTOK, CATOM);
    tokrange_kernel<<<(NTOK + 255) / 256, 256, 0, stream>>>(tok, ranges);
    segmean_kernel<<<(NTOK * CTOK + 255) / 256, 256, 0, stream>>>(pqws, ranges, outAI);
}